// VectorQuantizer_41609643164281
// MI455X (gfx1250) — compile-verified
//
#include <hip/hip_runtime.h>
#include <hip/hip_bf16.h>
#include <math.h>

typedef float v2f __attribute__((ext_vector_type(2)));
typedef float v8f __attribute__((ext_vector_type(8)));

#define NUM_EMB 512
#define EMB_DIM 64
#define N_TOT   65536      // 16*64*64 rows
#define HW      4096
#define CHW     262144     // 64*4096
#define ROWS    128        // rows per workgroup
#define KCHUNK  256        // codebook rows per LDS chunk
#define PITCH   66         // LDS pitch (bank-conflict free, float2-aligned)

// flat output offsets (floats): loss | quantized BCHW | perplexity | encodings
#define OUT_LOSS 0
#define OUT_Q    1
#define OUT_PERP 4194305
#define OUT_ENC  4194306L

__global__ __launch_bounds__(512) void vq_init(float* sse, unsigned int* counts) {
  if (threadIdx.x == 0) *sse = 0.f;
  if (threadIdx.x < NUM_EMB) counts[threadIdx.x] = 0u;
}

__global__ __launch_bounds__(256) void vq_main(const float* __restrict__ x,
                                               const float* __restrict__ emb,
                                               float* __restrict__ out,
                                               float* __restrict__ sse_acc,
                                               unsigned int* __restrict__ counts) {
  __shared__ float A[ROWS][PITCH];     // x tile, [row][c]
  __shared__ float E[KCHUNK][PITCH];   // codebook chunk, [k][c]
  __shared__ float enorm[KCHUNK];
  __shared__ int   idx_lds[ROWS];
  __shared__ float red[8];

  const int tid  = threadIdx.x;
  const int lane = tid & 31;
  const int wid  = tid >> 5;
  const int lm   = lane & 15;
  const int hi   = lane >> 4;
  const int n0   = blockIdx.x * ROWS;
  const int b    = n0 >> 12;           // batch
  const int hw0  = n0 & (HW - 1);      // hw offset inside plane

  // ---- load A tile: per c-plane, 128 consecutive floats (coalesced) ----
  for (int i = tid; i < ROWS * EMB_DIM; i += 256) {
    int c = i >> 7;
    int r = i & (ROWS - 1);
    A[r][c] = x[(long)b * CHW + (long)c * HW + (hw0 + r)];
  }
  __syncthreads();

  // ---- per-wave A fragments (16 rows x 64 c); ISA f32 A layout:
  //      V0: lanes0-15 K=0 / lanes16-31 K=2 ; V1: K=1 / K=3 ----
  v2f afrag[16];
  {
    int arow = wid * 16 + lm;
    #pragma unroll
    for (int d = 0; d < 16; ++d) {
      int c = 4 * d + 2 * hi;
      afrag[d].x = A[arow][c];
      afrag[d].y = A[arow][c + 1];
    }
  }

  float best[8];
  int   bestk[8];
  #pragma unroll
  for (int r = 0; r < 8; ++r) { best[r] = 3.4e38f; bestk[r] = 0; }

  for (int chunk = 0; chunk < NUM_EMB / KCHUNK; ++chunk) {
    __syncthreads();  // previous chunk fully consumed
    for (int i = tid; i < KCHUNK * EMB_DIM; i += 256) {
      int kl = i >> 6;
      int c  = i & 63;
      E[kl][c] = emb[(long)(chunk * KCHUNK + kl) * EMB_DIM + c];
    }
    __syncthreads();
    {
      float s = 0.f;
      #pragma unroll 8
      for (int c = 0; c < EMB_DIM; ++c) { float e = E[tid][c]; s += e * e; }
      enorm[tid] = s;
    }
    __syncthreads();

    for (int kt = 0; kt < KCHUNK / 16; ++kt) {
      v8f acc = {0.f, 0.f, 0.f, 0.f, 0.f, 0.f, 0.f, 0.f};
      int kl = kt * 16 + lm;
      #pragma unroll
      for (int d = 0; d < 16; ++d) {
        int c = 4 * d + 2 * hi;
        v2f bfrag;
        bfrag.x = E[kl][c];
        bfrag.y = E[kl][c + 1];
        // D = A(16x4 f32) * B(4x16 f32) + C ; exact fp32 for argmin fidelity
        acc = __builtin_amdgcn_wmma_f32_16x16x4_f32(false, afrag[d], false, bfrag,
                                                    (short)0, acc, false, false);
      }
      float en = enorm[kl];
      int   kg = chunk * KCHUNK + kt * 16 + lm;
      #pragma unroll
      for (int r = 0; r < 8; ++r) {
        float v = fmaf(-2.f, acc[r], en);   // ||e||^2 - 2 x.e (||x||^2 const per row)
        if (v < best[r]) { best[r] = v; bestk[r] = kg; }  // strict <: first-min wins
      }
    }
  }

  // ---- argmin across the 16 lanes holding each row (xor stays in lane-half) ----
  #pragma unroll
  for (int m = 1; m <= 8; m <<= 1) {
    #pragma unroll
    for (int r = 0; r < 8; ++r) {
      float ov = __shfl_xor(best[r], m, 32);
      int   ok = __shfl_xor(bestk[r], m, 32);
      if (ov < best[r] || (ov == best[r] && ok < bestk[r])) { best[r] = ov; bestk[r] = ok; }
    }
  }
  if (lm == 0) {
    #pragma unroll
    for (int r = 0; r < 8; ++r) idx_lds[wid * 16 + r + 8 * hi] = bestk[r];
  }
  __syncthreads();

  // ---- encodings one-hot: contiguous 256 KB block per WG, float2 stores ----
  {
    float2* enc = (float2*)(out + OUT_ENC + (long)n0 * NUM_EMB);
    for (int i = tid; i < ROWS * (NUM_EMB / 2); i += 256) {
      int r  = i >> 8;             // 256 float2 per row
      int c2 = (i & 255) << 1;
      int id = idx_lds[r];
      float2 v;
      v.x = (c2     == id) ? 1.f : 0.f;
      v.y = (c2 + 1 == id) ? 1.f : 0.f;
      enc[i] = v;
    }
  }

  // ---- quantized (BCHW, coalesced per c-plane) + SSE partial ----
  float sse = 0.f;
  for (int i = tid; i < EMB_DIM * ROWS; i += 256) {
    int c = i >> 7;
    int r = i & (ROWS - 1);
    float q  = emb[(long)idx_lds[r] * EMB_DIM + c];  // codebook hot in L2
    float xv = A[r][c];
    float dd = q - xv;
    sse = fmaf(dd, dd, sse);
    out[OUT_Q + (long)b * CHW + (long)c * HW + (hw0 + r)] = q;
  }
  #pragma unroll
  for (int m = 16; m >= 1; m >>= 1) sse += __shfl_xor(sse, m, 32);
  if (lane == 0) red[wid] = sse;
  __syncthreads();
  if (tid == 0) {
    float s = 0.f;
    #pragma unroll
    for (int w = 0; w < 8; ++w) s += red[w];
    atomicAdd(sse_acc, s);
  }

  // ---- histogram for perplexity ----
  if (tid < ROWS) atomicAdd(&counts[idx_lds[tid]], 1u);
}

__global__ __launch_bounds__(512) void vq_final(const float* __restrict__ sse,
                                                const unsigned int* __restrict__ counts,
                                                float* __restrict__ out) {
  __shared__ float red[16];
  int tid = threadIdx.x;
  float p    = (float)counts[tid] * (1.f / (float)N_TOT);
  float term = p * logf(p + 1e-10f);
  #pragma unroll
  for (int m = 16; m >= 1; m >>= 1) term += __shfl_xor(term, m, 32);
  if ((tid & 31) == 0) red[tid >> 5] = term;
  __syncthreads();
  if (tid == 0) {
    float H = 0.f;
    #pragma unroll
    for (int w = 0; w < 16; ++w) H += red[w];
    out[OUT_PERP] = expf(-H);
    out[OUT_LOSS] = 1.25f * (*sse) / (float)((long)N_TOT * EMB_DIM);
  }
}

extern "C" void kernel_launch(void* const* d_in, const int* in_sizes, int n_in,
                              void* d_out, int out_size, void* d_ws, size_t ws_size,
                              hipStream_t stream) {
  const float* x   = (const float*)d_in[0];   // [16,64,64,64] BCHW fp32
  const float* emb = (const float*)d_in[1];   // [512,64] fp32
  float* out = (float*)d_out;
  float* sse = (float*)d_ws;
  unsigned int* counts = (unsigned int*)((char*)d_ws + 256);

  vq_init<<<1, 512, 0, stream>>>(sse, counts);
  vq_main<<<N_TOT / ROWS, 256, 0, stream>>>(x, emb, out, sse, counts);
  vq_final<<<1, 512, 0, stream>>>(sse, counts, out);
}